// OrthogonalButterfly_54030688583985
// MI455X (gfx1250) — compile-verified
//
#include <hip/hip_runtime.h>
#include <stdint.h>
#include <math.h>

#define WIDTH 1024
#define HALF  512
#define DEPTH 20
#define TC    16     // batch columns per workgroup tile
#define LSTR  20     // LDS row pitch in floats (80B: 16B-aligned, bank-spread)
#define BLK   256    // 8 waves (wave32)

typedef __attribute__((ext_vector_type(2))) float v2f;
typedef __attribute__((ext_vector_type(8))) float v8f;

// ---------------- CDNA5 async global<->LDS helpers (ASYNCcnt path) ----------------
__device__ __forceinline__ void async_ld_b128(unsigned lds_off, unsigned long long gbase, unsigned voff) {
    asm volatile("global_load_async_to_lds_b128 %0, %1, %2"
                 :: "v"(lds_off), "v"(voff), "s"(gbase) : "memory");
}
__device__ __forceinline__ void async_st_b128(unsigned long long gbase, unsigned voff, unsigned lds_off) {
    asm volatile("global_store_async_from_lds_b128 %0, %1, %2"
                 :: "v"(voff), "v"(lds_off), "s"(gbase) : "memory");
}
__device__ __forceinline__ void wait_async0() {
    asm volatile("s_wait_asynccnt 0x0" ::: "memory");
}

// ---------------- prep: cos/sin tables for all 20 layers ----------------
__global__ __launch_bounds__(BLK) void trig_kernel(const float* __restrict__ angles,
                                                   float* __restrict__ cosT,
                                                   float* __restrict__ sinT) {
    int t = blockIdx.x * BLK + threadIdx.x;
    if (t < DEPTH * HALF) {
        float s, c;
        sincosf(angles[t], &s, &c);
        cosT[t] = c;
        sinT[t] = s;
    }
}

// ---------------- prep: fuse stride {1,2,4,8} layers into 16x16 block matrices ----
// mats[sweep][G][r][c]  (sweep in {0,1} -> layers sweep*10 + 0..3), 128 matrices.
__global__ __launch_bounds__(BLK) void mats_kernel(const float* __restrict__ cosT,
                                                   const float* __restrict__ sinT,
                                                   float* __restrict__ mats) {
    int t = blockIdx.x * BLK + threadIdx.x;   // 2*64*16 = 2048 threads: one column each
    if (t >= 2 * 64 * 16) return;
    int sweep = t >> 10;
    int rem   = t & 1023;
    int G     = rem >> 4;     // row-group 0..63
    int c     = rem & 15;     // column of the 16x16 block
    float v[16];
#pragma unroll
    for (int r = 0; r < 16; ++r) v[r] = (r == c) ? 1.0f : 0.0f;
#pragma unroll
    for (int k = 0; k < 4; ++k) {             // strides 1,2,4,8 (within 16-row group)
        const int s = 1 << k;
        const int l = sweep * 10 + k;
#pragma unroll
        for (int pp = 0; pp < 8; ++pp) {      // 8 pairs per 16-row group
            int q   = pp >> k;
            int tt  = pp & (s - 1);
            int r0  = (q << (k + 1)) + tt;
            int r1  = r0 + s;
            int gr0 = G * 16 + r0;
            int idx = ((gr0 >> (k + 1)) << k) + (gr0 & (s - 1));  // == global pair index
            float cv = cosT[l * HALF + idx];
            float sv = sinT[l * HALF + idx];
            float x0 = v[r0], x1 = v[r1];
            v[r0] = cv * x0 + sv * x1;
            v[r1] = cv * x1 - sv * x0;
        }
    }
    float* M = mats + (sweep * 64 + G) * 256;
#pragma unroll
    for (int r = 0; r < 16; ++r) M[r * 16 + c] = v[r];
}

// ---------------- main fused butterfly kernel ----------------
// One workgroup owns a 1024-row x 16-col slab in LDS.
// Per sweep: 4 low-stride layers as 16x16 block-diag WMMA (in place, one group at
// a time -- rows of group G are touched only by G's owning wave, so no cross-wave
// hazard and only one v8f accumulator live), then 6 elementwise layers.
__global__ __launch_bounds__(BLK) void butterfly_kernel(const float* __restrict__ X,
                                                        const float* __restrict__ cosT,
                                                        const float* __restrict__ sinT,
                                                        const float* __restrict__ mats,
                                                        float* __restrict__ out,
                                                        int batch) {
    extern __shared__ float lds[];            // WIDTH * LSTR floats = 80 KB
    const int tid = threadIdx.x;
    const int ct  = blockIdx.x;               // column-tile index
    const unsigned rowPitchB = (unsigned)batch * 4u;
    const unsigned colBaseB  = (unsigned)ct * (TC * 4u);

    // ---- async global -> LDS tile fill (1024 rows x 16 cols, B128 chunks) ----
    {
        unsigned long long xb = (unsigned long long)(uintptr_t)X;
#pragma unroll
        for (int kk = 0; kk < 16; ++kk) {
            int ch  = kk * BLK + tid;         // 4096 16-byte chunks
            int row = ch >> 2, c4 = ch & 3;
            unsigned loff = (unsigned)(uintptr_t)&lds[row * LSTR + c4 * 4];
            unsigned goff = (unsigned)row * rowPitchB + colBaseB + (unsigned)c4 * 16u;
            async_ld_b128(loff, xb, goff);
        }
        wait_async0();
        __syncthreads();
    }

    const int wave   = tid >> 5;
    const int lane16 = tid & 15;
    const int hi     = (tid >> 4) & 1;        // lane >= 16 within the wave32

    for (int sweep = 0; sweep < 2; ++sweep) {
        // ---- WMMA stage: rows[16G..16G+15] = M_G (16x16) * rows, K chained 4x4 ----
#pragma unroll 2
        for (int i = 0; i < 8; ++i) {
            const int G = wave * 8 + i;
            const float* M = mats + (sweep * 64 + G) * 256;
            v8f d = {0.f, 0.f, 0.f, 0.f, 0.f, 0.f, 0.f, 0.f};
#pragma unroll
            for (int j = 0; j < 4; ++j) {
                v2f av, bv;
                // A 16x4 layout: lanes 0-15 K={0,1}, lanes 16-31 K={2,3}, row = lane16
                av.x = M[lane16 * 16 + 4 * j + 2 * hi + 0];
                av.y = M[lane16 * 16 + 4 * j + 2 * hi + 1];
                // B 4x16 layout: VGPR0 rows K={0|2}, VGPR1 rows K={1|3}, col = lane16
                bv.x = lds[(G * 16 + 4 * j + 2 * hi + 0) * LSTR + lane16];
                bv.y = lds[(G * 16 + 4 * j + 2 * hi + 1) * LSTR + lane16];
                d = __builtin_amdgcn_wmma_f32_16x16x4_f32(
                        /*neg_a=*/false, av, /*neg_b=*/false, bv,
                        /*c_mod=*/(short)0, d, /*reuse_a=*/false, /*reuse_b=*/false);
            }
            // D layout: row = v + 8*hi, col = lane16; in-place writeback (same wave
            // already consumed these rows as B, per-wave DS ordering keeps RAW safe)
#pragma unroll
            for (int v = 0; v < 8; ++v)
                lds[(G * 16 + v + 8 * hi) * LSTR + lane16] = d[v];
        }
        __syncthreads();

        // ---- elementwise layers: strides 16,32,...,512 (disjoint pairs per layer) ----
        for (int k = 4; k < 10; ++k) {
            const int l = sweep * 10 + k;
            const float* cL = cosT + l * HALF;
            const float* sL = sinT + l * HALF;
#pragma unroll 4
            for (int it = 0; it < 32; ++it) {
                int task = it * BLK + tid;    // 512 pairs x 16 cols
                int c  = task & 15;
                int pp = task >> 4;           // angle index == pair index
                int r0 = (((pp >> k) << (k + 1))) + (pp & ((1 << k) - 1));
                int r1 = r0 + (1 << k);
                float cv = cL[pp], sv = sL[pp];
                float x0 = lds[r0 * LSTR + c];
                float x1 = lds[r1 * LSTR + c];
                lds[r0 * LSTR + c] = cv * x0 + sv * x1;
                lds[r1 * LSTR + c] = cv * x1 - sv * x0;
            }
            __syncthreads();
        }
    }

    // ---- async LDS -> global store of the finished tile ----
    {
        unsigned long long ob = (unsigned long long)(uintptr_t)out;
#pragma unroll
        for (int kk = 0; kk < 16; ++kk) {
            int ch  = kk * BLK + tid;
            int row = ch >> 2, c4 = ch & 3;
            unsigned loff = (unsigned)(uintptr_t)&lds[row * LSTR + c4 * 4];
            unsigned goff = (unsigned)row * rowPitchB + colBaseB + (unsigned)c4 * 16u;
            async_st_b128(ob, goff, loff);
        }
        wait_async0();
    }
}

extern "C" void kernel_launch(void* const* d_in, const int* in_sizes, int n_in,
                              void* d_out, int out_size, void* d_ws, size_t ws_size,
                              hipStream_t stream) {
    (void)n_in; (void)out_size; (void)ws_size;
    const float* X      = (const float*)d_in[0];   // (1024, 8192) f32
    const float* angles = (const float*)d_in[1];   // (20, 512)   f32
    float* out = (float*)d_out;

    float* cosT = (float*)d_ws;                    // 20*512 floats
    float* sinT = cosT + DEPTH * HALF;             // 20*512 floats
    float* mats = sinT + DEPTH * HALF;             // 2*64*256 floats (128 KB)

    int batch = in_sizes[0] / WIDTH;               // 8192

    trig_kernel<<<(DEPTH * HALF + BLK - 1) / BLK, BLK, 0, stream>>>(angles, cosT, sinT);
    mats_kernel<<<(2 * 64 * 16 + BLK - 1) / BLK, BLK, 0, stream>>>(cosT, sinT, mats);

    size_t shmem = (size_t)WIDTH * LSTR * sizeof(float);   // 80 KB
    butterfly_kernel<<<batch / TC, BLK, shmem, stream>>>(X, cosT, sinT, mats, out, batch);
}